// FeaStConv_51402168599240
// MI455X (gfx1250) — compile-verified
//
#include <hip/hip_runtime.h>

typedef float v2f __attribute__((ext_vector_type(2)));
typedef float v8f __attribute__((ext_vector_type(8)));

// FeaStConv constants: B=4, N=25000, F=64, K(neighbors)=H(heads)=9, C=32
constexpr int Fd = 64;
constexpr int KN = 9;     // neighbors
constexpr int H  = 9;     // heads
constexpr int C  = 32;    // output channels
constexpr int TM = 16;    // points per tile (WMMA M)
constexpr int KH = Fd * H;        // 576 : WMMA reduction extent (f,h) flattened

// LDS padding strides (bank-conflict avoidance across the 64 banks)
constexpr int XS_KS = 68;          // per-neighbor stride in xs (>= 64, 16B aligned)
constexpr int XS_MS = KN * XS_KS;  // 612 per-point stride (2448 B, 36*m mod 64 distinct)
constexpr int AT_MS = 89;          // attn per-point stride (>= 81, odd -> distinct banks)
constexpr int GS_MS = 581;         // G per-point stride    (>= 576, odd)
constexpr int WO_CS = 32;          // Wo row stride: unpadded so rows stay 16B-aligned
                                   // for async B128 staging (2-way B-read conflict ok)

__device__ __forceinline__ unsigned lds_off(const void* p) {
    // generic LDS pointer = {shared-aperture hi, LDS byte offset lo} -> truncate
    return (unsigned)(uintptr_t)p;
}

__device__ __forceinline__ void async_g2l_b128(unsigned lds_byte, unsigned gbyte,
                                               const void* sbase) {
    asm volatile("global_load_async_to_lds_b128 %0, %1, %2 offset:0"
                 :: "v"(lds_byte), "v"(gbyte), "s"(sbase) : "memory");
}

__global__ __launch_bounds__(64) void feast_conv_wmma(
    const float* __restrict__ x,          // [B*N, F]
    const long long* __restrict__ nb,     // [B*N, K] (indices within batch)
    const float* __restrict__ Wmlp,       // [F, H]
    const float* __restrict__ bmlp,       // [H]
    const float* __restrict__ Wout,       // [F, H*C]  (== [fh, c])
    const float* __restrict__ bias,       // [C]
    float* __restrict__ out,              // [B*N, C]
    int N, int BN)
{
    __shared__ float wo[KH * WO_CS];      // 73,728 B
    __shared__ float wmlp_s[Fd * H];      //  2,304 B
    __shared__ float bml[16];
    __shared__ float bsc[C];
    __shared__ float xs[TM * XS_MS];      // 39,168 B : gathered neighbor features
    __shared__ float at[TM * AT_MS];      //  5,696 B : softmaxed attention
    __shared__ float gs[TM * GS_MS];      // 37,184 B : G[m, f*H+h]
    __shared__ int   nbi[TM * KN];        //    576 B : resolved global row ids

    const int t    = threadIdx.x;
    const int tile = blockIdx.x;
    const int p0   = tile * TM;

    // ---- P0: async-stage weights into LDS (memory -> LDS, ASYNCcnt) ----
    {
        const unsigned woL = lds_off(&wo[0]);
        for (int i = t; i < (KH * C) / 4; i += 64)          // 4608 float4
            async_g2l_b128(woL + i * 16u, (unsigned)i * 16u, Wout);
        const unsigned wmL = lds_off(&wmlp_s[0]);
        for (int i = t; i < (Fd * H) / 4; i += 64)          // 144 float4
            async_g2l_b128(wmL + i * 16u, (unsigned)i * 16u, Wmlp);
    }
    if (t < H) bml[t] = bmlp[t];
    if (t < C) bsc[t] = bias[t];

    // ---- P1: resolve neighbor indices -> global rows ----
    for (int r = t; r < TM * KN; r += 64) {
        int m = r / KN, k = r - m * KN;
        int g = p0 + m;
        if (g < BN) {
            int b = g / N;
            long long j = nb[(long long)g * KN + k];
            nbi[r] = b * N + (int)j;
        } else {
            nbi[r] = 0;
        }
    }
    __syncthreads();

    // ---- P2: async gather of 16x9x64 neighbor tile (B128, L2-resident) ----
    for (int i = t; i < TM * KN * (Fd / 4); i += 64) {
        int row = i >> 4;                 // Fd/4 == 16 float4 per row
        int v   = i & 15;
        int m = row / KN, k = row - m * KN;
        unsigned gbyte = ((unsigned)nbi[row] * (unsigned)Fd + (unsigned)(v * 4)) * 4u;
        unsigned lbyte = lds_off(&xs[m * XS_MS + k * XS_KS + v * 4]);
        async_g2l_b128(lbyte, gbyte, x);
    }
    asm volatile("s_wait_asynccnt 0x0" ::: "memory");
    __syncthreads();

    // ---- P3: logits = (x_nb - x_nb[k=0]) . Wmlp + b ; softmax over H ----
    for (int r = t; r < TM * KN; r += 64) {
        int m = r / KN, k = r - m * KN;
        const float* xk = &xs[m * XS_MS + k * XS_KS];
        const float* x0 = &xs[m * XS_MS];
        float acc[H];
        #pragma unroll
        for (int h = 0; h < H; ++h) acc[h] = bml[h];
        for (int f = 0; f < Fd; ++f) {
            float d = xk[f] - x0[f];
            #pragma unroll
            for (int h = 0; h < H; ++h) acc[h] = fmaf(d, wmlp_s[f * H + h], acc[h]);
        }
        float mx = acc[0];
        #pragma unroll
        for (int h = 1; h < H; ++h) mx = fmaxf(mx, acc[h]);
        float e[H]; float sum = 0.f;
        #pragma unroll
        for (int h = 0; h < H; ++h) { e[h] = __expf(acc[h] - mx); sum += e[h]; }
        float inv = 1.f / sum;
        #pragma unroll
        for (int h = 0; h < H; ++h) at[m * AT_MS + k * H + h] = e[h] * inv;
    }
    __syncthreads();

    // ---- P4: G[m, f*H+h] = sum_k x_nb[m,k,f] * attn[m,k,h] ----
    {
        const int m  = t & 15;
        const int fq = t >> 4;            // 4 lanes per point, 16 f-values each
        float atr[KN * H];                // 81 regs, fully unrolled indices
        #pragma unroll
        for (int i = 0; i < KN * H; ++i) atr[i] = at[m * AT_MS + i];
        for (int fi = 0; fi < 16; ++fi) {
            int f = fq * 16 + fi;
            float xv[KN];
            #pragma unroll
            for (int k = 0; k < KN; ++k) xv[k] = xs[m * XS_MS + k * XS_KS + f];
            #pragma unroll
            for (int h = 0; h < H; ++h) {
                float s = 0.f;
                #pragma unroll
                for (int k = 0; k < KN; ++k) s = fmaf(xv[k], atr[k * H + h], s);
                gs[m * GS_MS + f * H + h] = s;
            }
        }
    }
    __syncthreads();

    // ---- P5: [16 x 576] x [576 x 32] via chained V_WMMA_F32_16X16X4_F32 ----
    // wave 0 -> channels 0..15, wave 1 -> channels 16..31
    {
        const int lane = t & 31;
        const int wav  = t >> 5;
        const int mA   = lane & 15;          // A-matrix row (16x4: M in lane%16)
        const int kO   = (lane >> 4) * 2;    // K pair: lanes 16-31 hold K=2,3
        const int col  = (lane & 15) + wav * 16;  // B column = output channel
        v8f acc = {0.f, 0.f, 0.f, 0.f, 0.f, 0.f, 0.f, 0.f};
        const float* gRow = &gs[mA * GS_MS];
        for (int kk = 0; kk < KH / 4; ++kk) {
            int fh = kk * 4 + kO;
            v2f a, b;
            a.x = gRow[fh];
            a.y = gRow[fh + 1];
            b.x = wo[fh * WO_CS + col];
            b.y = wo[(fh + 1) * WO_CS + col];
            acc = __builtin_amdgcn_wmma_f32_16x16x4_f32(
                false, a, false, b, (short)0, acc, false, false);
        }

        // ---- P6: epilogue per documented 16x16 f32 D layout ----
        const int mBase = (lane >> 4) * 8;   // VGPR r -> M = r + 8*(lane/16)
        #pragma unroll
        for (int r = 0; r < 8; ++r) {
            int m = mBase + r;
            int g = p0 + m;
            if (g < BN) {
                int b = g / N;
                int n = g - b * N;
                float v = acc[r] + bsc[col];
                v = fmaxf(v, 0.f);
                if (n == N - 1) v = 0.f;     // zero_padding on last vertex
                out[(long long)g * C + col] = v;
            }
        }
    }
}

extern "C" void kernel_launch(void* const* d_in, const int* in_sizes, int n_in,
                              void* d_out, int out_size, void* d_ws, size_t ws_size,
                              hipStream_t stream) {
    (void)n_in; (void)out_size; (void)d_ws; (void)ws_size;
    const float*     x    = (const float*)d_in[0];
    // d_in[1] = t_vertex (unused by forward)
    const long long* nb   = (const long long*)d_in[2];   // int64 neighbor_index
    const float*     Wm   = (const float*)d_in[3];
    const float*     bm   = (const float*)d_in[4];
    const float*     Wo   = (const float*)d_in[5];
    const float*     bs   = (const float*)d_in[6];
    float*           outp = (float*)d_out;

    const int BN = in_sizes[0] / Fd;   // B*N = 100000
    const int B  = 4;
    const int N  = BN / B;             // 25000
    const int tiles = (BN + TM - 1) / TM;

    feast_conv_wmma<<<tiles, 64, 0, stream>>>(x, nb, Wm, bm, Wo, bs, outp, N, BN);
}